// SingleValueAgent_11639361372461
// MI455X (gfx1250) — compile-verified
//
#include <hip/hip_runtime.h>

// EWMA scan as tiled lower-triangular matmul on CDNA5 WMMA (f32 16x16x4).
// out[s,t] = beta*out[s,t-1] + alpha*x0*x1 ; alpha = sigmoid(a_raw), beta = 1-alpha
//
// Tile view (16 trials x 16 sessions):  D = L * X + (beta^(t+1)) (x) carry
//   L[t,k] = alpha * beta^(t-k)  (k<=t, else 0)   -- constant, built once per lane
//   X[k,n] = x[s0+n, t0+k, 0] * x[s0+n, t0+k, 1]
// One wave per 16-session group, 256 sequential tiles.
// 8-deep branch-free load pipeline; last 8 tiles peeled into an epilogue so the
// steady-state loop has no conditional (avoids phi-copy register churn).

typedef __attribute__((ext_vector_type(2))) float v2f;
typedef __attribute__((ext_vector_type(4))) float v4f;
typedef __attribute__((ext_vector_type(8))) float v8f;

#define N_SESS   8192
#define N_TRIALS 4096
#define TILE     16
#define ITERS    (N_TRIALS / TILE)   // 256
#define PIPE     8                   // load pipeline depth (256 % 8 == 0)

__device__ __forceinline__ float powi(float b, int e) {
    float p = 1.0f;
    for (int i = 0; i < e; ++i) p *= b;   // e <= 16, setup-only
    return p;
}

__global__ __launch_bounds__(256) void ewma_scan_wmma(const float* __restrict__ x,
                                                      const float* __restrict__ a_raw,
                                                      float* __restrict__ out) {
    const int lane = threadIdx.x & 31;
    const int wave = (blockIdx.x * blockDim.x + threadIdx.x) >> 5;
    const int n    = lane & 15;   // session column (B/C/D N-index) == A-matrix row M
    const int h    = lane >> 4;   // lane half
    const int sess = wave * 16 + n;

    // alpha = sigmoid(a_raw) (uniform)
    const float a     = a_raw[0];
    const float alpha = 1.0f / (1.0f + __expf(-a));
    const float beta  = 1.0f - alpha;

    // A-operand tiles: chunk c covers K = 4c..4c+3.
    // f32 16x4 A layout: lane row m = lane&15; VGPR0 holds K=2h, VGPR1 holds K=2h+1.
    const int m = n;
    v2f A[4];
#pragma unroll
    for (int c = 0; c < 4; ++c) {
        const int k0 = 4 * c + 2 * h;
        const int k1 = k0 + 1;
        A[c].x = (k0 <= m) ? alpha * powi(beta, m - k0) : 0.0f;
        A[c].y = (k1 <= m) ? alpha * powi(beta, m - k1) : 0.0f;
    }

    // Carry decay per accumulator row: acc[r] holds trial M = r + 8h -> beta^(M+1)
    float pw[8];
#pragma unroll
    for (int r = 0; r < 8; ++r) pw[r] = powi(beta, r + 8 * h + 1);

    const v4f* px = reinterpret_cast<const v4f*>(x + (size_t)sess * (2 * N_TRIALS));
    float*     ob = out + (size_t)sess * N_TRIALS + 8 * h;

    // Deep pipeline: each b128 grabs trials (t0+4c+2h, t0+4c+2h+1) for session n,
    // i.e. exactly the two K slots this lane owns in B-chunk c.
    v4f buf[PIPE][4];
#pragma unroll
    for (int p = 0; p < PIPE; ++p)
#pragma unroll
        for (int c = 0; c < 4; ++c)
            buf[p][c] = __builtin_nontemporal_load(&px[(p * TILE + 4 * c + 2 * h) >> 1]);

    v8f acc = {0.f, 0.f, 0.f, 0.f, 0.f, 0.f, 0.f, 0.f};  // state_{-1} = 0

    // One tile step: consume b (tile `it`), optionally prefetch tile it+PIPE into b.
    auto tile_step = [&](int it, v4f (&b)[4], bool doLoad) {
        const int t0 = it * TILE;

        // B chunks: xt = x0*x1 for this lane's two K slots
        v2f B[4];
#pragma unroll
        for (int c = 0; c < 4; ++c) {
            const v4f f = b[c];
            B[c].x = f.x * f.y;
            B[c].y = f.z * f.w;
        }

        if (doLoad) {
#pragma unroll
            for (int c = 0; c < 4; ++c)
                b[c] = __builtin_nontemporal_load(
                    &px[((it + PIPE) * TILE + 4 * c + 2 * h) >> 1]);
        }

        // D = sum_c L[:,4c:4c+4] * X[4c:4c+4,:] + carry-seeded acc
#pragma unroll
        for (int c = 0; c < 4; ++c)
            acc = __builtin_amdgcn_wmma_f32_16x16x4_f32(
                false, A[c], false, B[c], (short)0, acc, false, false);

        // Store: acc[r] -> out[sess, t0 + 8h + r] (contiguous 8 trials per lane)
        const v4f lo = {acc[0], acc[1], acc[2], acc[3]};
        const v4f hi = {acc[4], acc[5], acc[6], acc[7]};
        v4f* po = reinterpret_cast<v4f*>(ob + t0);
        __builtin_nontemporal_store(lo, po);
        __builtin_nontemporal_store(hi, po + 1);

        // Carry = row M=15 (acc VGPR 7, lanes 16..31); broadcast to column owners
        const float carry = __shfl(acc[7], (lane & 15) + 16, 32);
#pragma unroll
        for (int r = 0; r < 8; ++r) acc[r] = pw[r] * carry;
    };

    // Steady state: always prefetch — branch-free bodies, one branch per PIPE tiles.
    for (int ot = 0; ot < ITERS - PIPE; ot += PIPE) {
#pragma unroll
        for (int j = 0; j < PIPE; ++j)
            tile_step(ot + j, buf[j], true);
    }

    // Epilogue: final PIPE tiles, no further loads.
#pragma unroll
    for (int j = 0; j < PIPE; ++j)
        tile_step(ITERS - PIPE + j, buf[j], false);
}

extern "C" void kernel_launch(void* const* d_in, const int* in_sizes, int n_in,
                              void* d_out, int out_size, void* d_ws, size_t ws_size,
                              hipStream_t stream) {
    (void)in_sizes; (void)n_in; (void)out_size; (void)d_ws; (void)ws_size;
    const float* x     = (const float*)d_in[0];
    const float* a_raw = (const float*)d_in[1];
    float*       out   = (float*)d_out;

    const int waves   = N_SESS / 16;          // 512 waves
    const int threads = waves * 32;           // 16384
    dim3 block(256);
    dim3 grid(threads / 256);                 // 64 blocks
    ewma_scan_wmma<<<grid, block, 0, stream>>>(x, a_raw, out);
}